// CompLayer_50448685859252
// MI455X (gfx1250) — compile-verified
//
#include <hip/hip_runtime.h>
#include <math.h>

// Problem constants (from reference)
#define N_ENT 150000
#define N_RELS 500
#define HDIM 128
#define DEG 32
#define TOPK 10
#define A_STRIDE 132   // padded LDS stride (128+4): avoids 16-way bank conflicts on A-column reads

typedef __attribute__((ext_vector_type(2))) float v2f;
typedef __attribute__((ext_vector_type(8))) float v8f;

__device__ __forceinline__ float wave_sum32(float v) {
#pragma unroll
  for (int off = 16; off > 0; off >>= 1) v += __shfl_xor(v, off, 32);
  return v;
}

// --------------------------------------------------------------------------
// One wave computes a 16(M) x 16(N) f32 tile of  D = A(16x128) * B(128x128)
// using V_WMMA_F32_16X16X4_F32 chained over K (32 steps).
// A layout (ISA 7.12.2, 32-bit A 16x4): VGPR0 = K{0,2}, VGPR1 = K{1,3};
// lanes 0-15 -> K base 0, lanes 16-31 -> K base +2. B/C/D: row-per-VGPR with
// rows r and r+half paired across lane halves.
// --------------------------------------------------------------------------
__device__ __forceinline__ v8f wmma_tile_16x128(const float* lds_a,
                                                const float* lds_b,
                                                int lane, int ntile) {
  v8f acc = {0.f, 0.f, 0.f, 0.f, 0.f, 0.f, 0.f, 0.f};
  const int m    = lane & 15;
  const int koff = (lane >> 4) << 1;          // 0 (lanes 0-15) or 2 (lanes 16-31)
  const int ncol = ntile * 16 + (lane & 15);
#pragma unroll 4
  for (int kb = 0; kb < HDIM; kb += 4) {
    const int k = kb + koff;
    v2f af, bf;
    af.x = lds_a[m * A_STRIDE + k];
    af.y = lds_a[m * A_STRIDE + k + 1];
    bf.x = lds_b[(k)*HDIM + ncol];
    bf.y = lds_b[(k + 1) * HDIM + ncol];
    acc = __builtin_amdgcn_wmma_f32_16x16x4_f32(false, af, false, bf,
                                                (short)0, acc, false, false);
  }
  return acc;
}

// --------------------------------------------------------------------------
// Kernel 1: h_rel = rel_emb @ W_r ; s_rel = h_rel @ a3   (500 rows, trivial)
// --------------------------------------------------------------------------
__global__ void rel_kernel(const float* __restrict__ rel_emb,
                           const float* __restrict__ W_r,
                           const float* __restrict__ a,
                           float* __restrict__ h_rel,
                           float* __restrict__ s_rel) {
  __shared__ float red[HDIM];
  const int r = blockIdx.x;
  const int n = threadIdx.x;
  float acc = 0.f;
  for (int k = 0; k < HDIM; ++k)
    acc = fmaf(rel_emb[r * HDIM + k], W_r[k * HDIM + n], acc);
  h_rel[r * HDIM + n] = acc;
  red[n] = acc * a[2 * HDIM + n];
  __syncthreads();
  for (int s = HDIM / 2; s > 0; s >>= 1) {
    if (n < s) red[n] += red[n + s];
    __syncthreads();
  }
  if (n == 0) s_rel[r] = red[0];
}

// --------------------------------------------------------------------------
// Kernel 2: h_node = ent_emb @ W.  256 threads = 8 waves; block handles 16
// entity rows; wave w computes N-tile w. W staged once in LDS (64 KB).
// --------------------------------------------------------------------------
__global__ void hnode_kernel(const float* __restrict__ ent_emb,
                             const float* __restrict__ W,
                             float* __restrict__ h_node) {
  extern __shared__ float smem[];
  float* lds_w = smem;                  // 128*128
  float* lds_a = smem + HDIM * HDIM;    // 16*A_STRIDE
  const int tid  = threadIdx.x;
  const int base = blockIdx.x * 16;

  {  // stage W: 4096 float4
    const float4* s4 = (const float4*)W;
    float4* d4 = (float4*)lds_w;
#pragma unroll
    for (int i = 0; i < 16; ++i) d4[tid + i * 256] = s4[tid + i * 256];
  }
  {  // stage A tile: 512 float4 (padded rows stay 16B aligned: 132 % 4 == 0)
#pragma unroll
    for (int i = 0; i < 2; ++i) {
      const int f = tid + i * 256;
      const int row = f >> 5;
      const int c4 = f & 31;
      float4 v = ((const float4*)(ent_emb + (size_t)(base + row) * HDIM))[c4];
      *(float4*)(lds_a + row * A_STRIDE + c4 * 4) = v;
    }
  }
  __syncthreads();

  const int wave = tid >> 5, lane = tid & 31;
  v8f acc = wmma_tile_16x128(lds_a, lds_w, lane, wave);
  const int ncol = wave * 16 + (lane & 15);
  const int radd = (lane >> 4) << 3;  // lanes 16-31 hold rows M+8
#pragma unroll
  for (int v = 0; v < 8; ++v)
    h_node[(size_t)(base + radd + v) * HDIM + ncol] = acc[v];
}

// --------------------------------------------------------------------------
// Kernel 3: s_src = h_node@a1, s_dst = h_node@a2.  One wave per entity.
// --------------------------------------------------------------------------
__global__ void score_kernel(const float* __restrict__ h_node,
                             const float* __restrict__ a,
                             float* __restrict__ s_src,
                             float* __restrict__ s_dst) {
  const int tid = threadIdx.x;
  const int wave = tid >> 5, lane = tid & 31;
  const int n = blockIdx.x * 8 + wave;
  float4 h  = ((const float4*)(h_node + (size_t)n * HDIM))[lane];
  float4 a1 = ((const float4*)(a))[lane];
  float4 a2 = ((const float4*)(a + HDIM))[lane];
  float p1 = h.x * a1.x + h.y * a1.y + h.z * a1.z + h.w * a1.w;
  float p2 = h.x * a2.x + h.y * a2.y + h.z * a2.z + h.w * a2.w;
  p1 = wave_sum32(p1);
  p2 = wave_sum32(p2);
  if (lane == 0) { s_src[n] = p1; s_dst[n] = p2; }
}

// --------------------------------------------------------------------------
// Kernel 4 (fused): per-entity scoring + top-10 + softmax + gather-reduce
// into a 16-row LDS tile, then out = tanh(tile @ neigh_w) via WMMA.
// 256 threads = 8 waves; wave w handles entities 2w, 2w+1 then N-tile w.
// --------------------------------------------------------------------------
__global__ void attn_out_kernel(const float* __restrict__ h_node,
                                const float* __restrict__ h_rel,
                                const float* __restrict__ s_src,
                                const float* __restrict__ s_dst,
                                const float* __restrict__ s_rel,
                                const int* __restrict__ src,
                                const int* __restrict__ rel_id,
                                const float* __restrict__ neigh_w,
                                float* __restrict__ out) {
  extern __shared__ float smem[];
  float* lds_w = smem;                  // 128*128 (neigh_w)
  float* lds_n = smem + HDIM * HDIM;    // 16*A_STRIDE (neigh tile)
  const int tid  = threadIdx.x;
  const int wave = tid >> 5, lane = tid & 31;
  const int base = blockIdx.x * 16;

  {  // stage neigh_w
    const float4* s4 = (const float4*)neigh_w;
    float4* d4 = (float4*)lds_w;
#pragma unroll
    for (int i = 0; i < 16; ++i) d4[tid + i * 256] = s4[tid + i * 256];
  }

  // ---- attention phase: one wave per entity (lane = neighbor slot) ----
  for (int e = 0; e < 2; ++e) {
    const int row = wave * 2 + e;
    const int n = base + row;
    const int j = src[(size_t)n * DEG + lane];
    const int r = rel_id[(size_t)n * DEG + lane];
    float sc = s_src[j] + s_dst[n] + s_rel[r];
    sc = (sc > 0.f) ? sc : 0.2f * sc;  // leaky_relu(0.2)

    // iterative top-10 by wave argmax (lower lane wins ties, like lax.top_k)
    float cur = sc;
    float selv = 0.f; int selj = 0, selr = 0;
#pragma unroll
    for (int k = 0; k < TOPK; ++k) {
      float v = cur; int ii = lane;
#pragma unroll
      for (int off = 16; off > 0; off >>= 1) {
        float ov = __shfl_xor(v, off, 32);
        int   oi = __shfl_xor(ii, off, 32);
        if (ov > v || (ov == v && oi < ii)) { v = ov; ii = oi; }
      }
      const int wj = __shfl(j, ii, 32);
      const int wr = __shfl(r, ii, 32);
      if (lane == k) { selv = v; selj = wj; selr = wr; }
      if (lane == ii) cur = -3.4e38f;  // remove winner
    }

    // softmax over the 10 selections (selection 0 is the max)
    const float maxv = __shfl(selv, 0, 32);
    float ex = (lane < TOPK) ? expf(selv - maxv) : 0.f;
    const float inv = 1.f / wave_sum32(ex);

    // gather + weighted sum; lane owns columns [4*lane, 4*lane+4)
    float acc0 = 0.f, acc1 = 0.f, acc2 = 0.f, acc3 = 0.f;
#pragma unroll
    for (int k = 0; k < TOPK; ++k) {
      const int jk = __shfl(selj, k, 32);
      const int rk = __shfl(selr, k, 32);
      const float ak = __shfl(ex, k, 32) * inv;
      float4 hn = ((const float4*)(h_node + (size_t)jk * HDIM))[lane];
      float4 hr = ((const float4*)(h_rel + (size_t)rk * HDIM))[lane];
      acc0 = fmaf(ak, hn.x + hr.x, acc0);
      acc1 = fmaf(ak, hn.y + hr.y, acc1);
      acc2 = fmaf(ak, hn.z + hr.z, acc2);
      acc3 = fmaf(ak, hn.w + hr.w, acc3);
    }
    float4 o = {acc0, acc1, acc2, acc3};
    *(float4*)(lds_n + row * A_STRIDE + lane * 4) = o;
  }
  __syncthreads();

  // ---- output GEMM phase: wave w -> N-tile w, then tanh ----
  v8f acc = wmma_tile_16x128(lds_n, lds_w, lane, wave);
  const int ncol = wave * 16 + (lane & 15);
  const int radd = (lane >> 4) << 3;
#pragma unroll
  for (int v = 0; v < 8; ++v)
    out[(size_t)(base + radd + v) * HDIM + ncol] = tanhf(acc[v]);
}

// --------------------------------------------------------------------------
extern "C" void kernel_launch(void* const* d_in, const int* in_sizes, int n_in,
                              void* d_out, int out_size, void* d_ws, size_t ws_size,
                              hipStream_t stream) {
  const float* ent_emb = (const float*)d_in[0];
  const float* rel_emb = (const float*)d_in[1];
  const float* W       = (const float*)d_in[2];
  const float* W_r     = (const float*)d_in[3];
  const float* a       = (const float*)d_in[4];
  const float* neigh_w = (const float*)d_in[5];
  const int*   src     = (const int*)d_in[6];
  const int*   rel_id  = (const int*)d_in[7];
  float* out = (float*)d_out;

  // workspace carve-up (~78.2 MB)
  char* ws = (char*)d_ws;
  float* h_node = (float*)ws; ws += (size_t)N_ENT * HDIM * sizeof(float);
  float* h_rel  = (float*)ws; ws += (size_t)N_RELS * HDIM * sizeof(float);
  float* s_src  = (float*)ws; ws += (size_t)N_ENT * sizeof(float);
  float* s_dst  = (float*)ws; ws += (size_t)N_ENT * sizeof(float);
  float* s_rel  = (float*)ws; ws += (size_t)N_RELS * sizeof(float);

  const size_t smem = (size_t)(HDIM * HDIM + 16 * A_STRIDE) * sizeof(float); // 73984 B

  hipLaunchKernelGGL(rel_kernel, dim3(N_RELS), dim3(HDIM), 0, stream,
                     rel_emb, W_r, a, h_rel, s_rel);
  hipLaunchKernelGGL(hnode_kernel, dim3(N_ENT / 16), dim3(256), smem, stream,
                     ent_emb, W, h_node);
  hipLaunchKernelGGL(score_kernel, dim3(N_ENT / 8), dim3(256), 0, stream,
                     h_node, a, s_src, s_dst);
  hipLaunchKernelGGL(attn_out_kernel, dim3(N_ENT / 16), dim3(256), smem, stream,
                     h_node, h_rel, s_src, s_dst, s_rel, src, rel_id, neigh_w, out);
}